// KBpA_50027779064417
// MI455X (gfx1250) — compile-verified
//
#include <hip/hip_runtime.h>

typedef __attribute__((ext_vector_type(2))) float v2f;
typedef __attribute__((ext_vector_type(8))) float v8f;

#define NKB 256              // K columns (betas rows)
#define NC  1024             // number of RBF centers
#define WAVES_PER_BLOCK 8
#define PIXELS_PER_WAVE 16
#define PIXELS_PER_BLOCK (WAVES_PER_BLOCK * PIXELS_PER_WAVE)   // 128

// out[i] = sum_j exp(-|| (pix_i - (K@beta)_i) - c_j ||^2) * alpha_j
// Distance tile via V_WMMA_F32_16X16X4_F32:
//   A (16x4) row M  = [dx, dy, |d|^2, 1]
//   B (4x16) col N  = [-2cx, -2cy, 1, |c|^2]
//   C[M][N] = |d_M - c_N|^2
__global__ __launch_bounds__(256)
void kbpa_wmma_kernel(const float* __restrict__ K,
                      const float* __restrict__ betas,
                      const float* __restrict__ pixels,
                      const float* __restrict__ centers,
                      const float* __restrict__ alphas,
                      float* __restrict__ out)
{
    __shared__ float4 cen4[NC];        // { -2cx, -2cy, cx^2+cy^2, alpha }
    __shared__ float4 bet4[NKB / 2];   // bet4[i] = { b[2i][0], b[2i][1], b[2i+1][0], b[2i+1][1] }

    const int tid = threadIdx.x;

    // ---- stage centers + alphas + betas into LDS (once per block) ----
    for (int j = tid; j < NC; j += 256) {
        float cx = centers[2 * j + 0];
        float cy = centers[2 * j + 1];
        cen4[j] = make_float4(-2.0f * cx, -2.0f * cy, cx * cx + cy * cy, alphas[j]);
    }
    if (tid < NKB / 2) {
        bet4[tid] = make_float4(betas[4 * tid + 0], betas[4 * tid + 1],
                                betas[4 * tid + 2], betas[4 * tid + 3]);
    }
    __syncthreads();

    const int lane = tid & 31;
    const int wave = tid >> 5;
    const int p    = lane & 15;   // pixel within wave / N column within tile
    const int h    = lane >> 4;   // lane half selects A/B K-columns {0,1} vs {2,3}
    const int pixel_base = (blockIdx.x * WAVES_PER_BLOCK + wave) * PIXELS_PER_WAVE;
    const int row = pixel_base + p;

    // ---- phase 1: deformation  d = pixel - K @ betas  (lane half splits k-range) ----
    const float4* Kr = (const float4*)(K + (size_t)row * NKB + (size_t)h * (NKB / 2));
    float sx = 0.0f, sy = 0.0f;
#pragma unroll 4
    for (int it = 0; it < NKB / 8; ++it) {           // 32 iters x 4 k-values
        float4 kq  = Kr[it];
        int    bi  = h * (NKB / 4) + it * 2;         // = k/2 with k = h*128 + it*4
        float4 b01 = bet4[bi];
        float4 b23 = bet4[bi + 1];
        sx += kq.x * b01.x + kq.y * b01.z + kq.z * b23.x + kq.w * b23.z;
        sy += kq.x * b01.y + kq.y * b01.w + kq.z * b23.y + kq.w * b23.w;
    }
    sx += __shfl_xor(sx, 16, 32);
    sy += __shfl_xor(sy, 16, 32);

    const float px = pixels[2 * row + 0];
    const float py = pixels[2 * row + 1];
    const float dx = px - sx;
    const float dy = py - sy;
    const float psq = dx * dx + dy * dy;

    // A operand, ISA layout for 16x4 f32: V0 = K0 | K2, V1 = K1 | K3 across lane halves
    v2f A;
    A.x = h ? psq  : dx;   // col 0 (low lanes) / col 2 (high lanes)
    A.y = h ? 1.0f : dy;   // col 1 (low lanes) / col 3 (high lanes)

    float acc[8];
#pragma unroll
    for (int m = 0; m < 8; ++m) acc[m] = 0.0f;

    const float NLOG2E = -1.44269504088896340736f;   // exp(-x) = exp2(-x*log2(e))

    // ---- phase 2: 64 tiles of 16 centers; WMMA -> exp -> alpha-weighted accumulate ----
    for (int tile = 0; tile < NC / 16; ++tile) {
        float4 cb = cen4[tile * 16 + p];
        // B operand mirrors A layout: V0 = K0 | K2, V1 = K1 | K3
        v2f B;
        B.x = h ? 1.0f : cb.x;   // row 0 (-2cx) / row 2 (1)
        B.y = h ? cb.z : cb.y;   // row 1 (-2cy) / row 3 (c^2)

        v8f c = {};
        c = __builtin_amdgcn_wmma_f32_16x16x4_f32(
                /*neg_a=*/false, A, /*neg_b=*/false, B,
                /*c_mod=*/(short)0, c, /*reuse_a=*/false, /*reuse_b=*/false);

        const float alpha = cb.w;
#pragma unroll
        for (int m = 0; m < 8; ++m)
            acc[m] += __builtin_amdgcn_exp2f(c[m] * NLOG2E) * alpha;
    }

    // ---- reduce over N (16 lanes per half); C layout: lane=N, vgpr m -> M = m + 8*h ----
#pragma unroll
    for (int m = 0; m < 8; ++m) {
        acc[m] += __shfl_xor(acc[m], 1, 32);
        acc[m] += __shfl_xor(acc[m], 2, 32);
        acc[m] += __shfl_xor(acc[m], 4, 32);
        acc[m] += __shfl_xor(acc[m], 8, 32);
    }

    float val = acc[0];
#pragma unroll
    for (int m = 1; m < 8; ++m) val = (p == m) ? acc[m] : val;
    if (p < 8) out[pixel_base + p + 8 * h] = val;
}

extern "C" void kernel_launch(void* const* d_in, const int* in_sizes, int n_in,
                              void* d_out, int out_size, void* d_ws, size_t ws_size,
                              hipStream_t stream) {
    const float* K       = (const float*)d_in[0];
    const float* betas   = (const float*)d_in[1];
    const float* pixels  = (const float*)d_in[2];
    const float* centers = (const float*)d_in[3];
    const float* alphas  = (const float*)d_in[4];
    float*       out     = (float*)d_out;

    const int n_pixels = in_sizes[2] / 2;              // (n_pixels, 2)
    const int blocks   = n_pixels / PIXELS_PER_BLOCK;  // 262144/128 = 2048

    kbpa_wmma_kernel<<<blocks, 256, 0, stream>>>(K, betas, pixels, centers, alphas, out);
}